// Decoder_40742059770033
// MI455X (gfx1250) — compile-verified
//
#include <hip/hip_runtime.h>
#include <math.h>

#define BATCH  8192
#define VOCAB  512
#define EMBED  50
#define HIDDEN 64
#define MAXLEN 64
#define ROWS   16     // batch rows per workgroup (one WMMA M-tile)
#define NW     8      // waves per workgroup (wave32)

typedef __attribute__((ext_vector_type(2))) float v2f;
typedef __attribute__((ext_vector_type(8))) float v8f;

__device__ __forceinline__ v8f wmma_f32(v2f a, v2f b, v8f c) {
  // D = A(16x4,f32) * B(4x16,f32) + C(16x16,f32)
  return __builtin_amdgcn_wmma_f32_16x16x4_f32(
      /*neg_a=*/false, a, /*neg_b=*/false, b,
      /*c_mod=*/(short)0, c, /*reuse_a=*/false, /*reuse_b=*/false);
}

__global__ __launch_bounds__(256) void Decoder_40742059770033_kernel(
    const float* __restrict__ enc_h,   // [1,B,H]
    const float* __restrict__ embed,   // [V,E]
    const float* __restrict__ w_ih,    // [3H,E]
    const float* __restrict__ w_hh,    // [3H,H]
    const float* __restrict__ b_ih,    // [3H]
    const float* __restrict__ b_hh,    // [3H]
    const float* __restrict__ fc_w,    // [V,H]
    const float* __restrict__ fc_b,    // [V]
    float* __restrict__ out_logp,      // [B,T,V]
    float* __restrict__ out_h)         // [B,H]
{
  __shared__ float xbuf[ROWS][64];          // embedded input, zero-padded K to 64
  __shared__ float hbuf[2][ROWS][HIDDEN];   // ping-pong hidden state
  __shared__ float gibuf[ROWS][192];        // x @ w_ih^T + b_ih
  __shared__ float ghbuf[ROWS][192];        // h @ w_hh^T + b_hh
  __shared__ float redv[NW][ROWS];          // per-wave reduction partials (value)
  __shared__ int   redi[NW][ROWS];          // per-wave reduction partials (index)
  __shared__ float rowmax[ROWS];
  __shared__ float rowls[ROWS];             // log(sum(exp))
  __shared__ int   ids[ROWS];               // greedy token feedback

  const int tid  = threadIdx.x;
  const int wave = tid >> 5;
  const int lane = tid & 31;
  const int l16  = lane & 15;
  const int hi   = (lane >> 4) & 1;         // which 16-lane half
  const int r0   = blockIdx.x * ROWS;

  // init: h0 from encoder hidden, ids = SOS
  for (int e = tid; e < ROWS * HIDDEN; e += 256) {
    int m = e >> 6, k = e & 63;
    hbuf[0][m][k] = enc_h[(size_t)(r0 + m) * HIDDEN + k];
  }
  if (tid < ROWS) ids[tid] = 1;  // SOS
  __syncthreads();

  int p = 0;
  for (int t = 0; t < MAXLEN; ++t) {
    // ---- embedding gather (zero pad K = 50..63) ----
    for (int e = tid; e < ROWS * 64; e += 256) {
      int m = e >> 6, k = e & 63;
      float v = 0.0f;
      if (k < EMBED) v = embed[ids[m] * EMBED + k];
      xbuf[m][k] = v;
    }
    __syncthreads();

    // ---- gate GEMMs: 12 N-tiles of 16 across 8 waves ----
    for (int tile = wave; tile < 12; tile += NW) {
      const int col = tile * 16 + l16;

      // gi = x @ w_ih^T + b_ih   (valid K = 50, A zero-padded)
      v8f ci;
      { float bi = b_ih[col];
        #pragma unroll
        for (int v = 0; v < 8; ++v) ci[v] = bi; }
      #pragma unroll
      for (int kk = 0; kk < 13; ++kk) {          // K = 0..51
        int k = kk * 4 + hi * 2;
        v2f a; a.x = xbuf[l16][k]; a.y = xbuf[l16][k + 1];
        v2f b;
        b.x = (k     < EMBED) ? w_ih[col * EMBED + k]     : 0.0f;
        b.y = (k + 1 < EMBED) ? w_ih[col * EMBED + k + 1] : 0.0f;
        ci = wmma_f32(a, b, ci);
      }

      // gh = h @ w_hh^T + b_hh   (K = 64)
      v8f ch;
      { float bh = b_hh[col];
        #pragma unroll
        for (int v = 0; v < 8; ++v) ch[v] = bh; }
      #pragma unroll
      for (int kk = 0; kk < 16; ++kk) {
        int k = kk * 4 + hi * 2;
        v2f a; a.x = hbuf[p][l16][k]; a.y = hbuf[p][l16][k + 1];
        v2f b; b.x = w_hh[col * HIDDEN + k]; b.y = w_hh[col * HIDDEN + k + 1];
        ch = wmma_f32(a, b, ch);
      }

      #pragma unroll
      for (int v = 0; v < 8; ++v) {
        int m = v + hi * 8;                 // C/D layout: VGPR v -> rows v / v+8
        gibuf[m][col] = ci[v];
        ghbuf[m][col] = ch[v];
      }
    }
    __syncthreads();

    // ---- GRU elementwise (torch semantics) ----
    for (int e = tid; e < ROWS * HIDDEN; e += 256) {
      int m = e >> 6, k = e & 63;
      float ir = gibuf[m][k],       hr = ghbuf[m][k];
      float iz = gibuf[m][64 + k],  hz = ghbuf[m][64 + k];
      float in_ = gibuf[m][128 + k], hn = ghbuf[m][128 + k];
      float r = 1.0f / (1.0f + __expf(-(ir + hr)));
      float z = 1.0f / (1.0f + __expf(-(iz + hz)));
      float n = tanhf(in_ + r * hn);
      hbuf[p ^ 1][m][k] = (1.0f - z) * n + z * hbuf[p][m][k];
    }
    __syncthreads();

    // ---- FC: logits = h_new @ fc_w^T + fc_b ; each wave owns 64 vocab cols ----
    v8f acc[4];
    #pragma unroll
    for (int tt = 0; tt < 4; ++tt) {
      int col = wave * 64 + tt * 16 + l16;
      float fb = fc_b[col];
      #pragma unroll
      for (int v = 0; v < 8; ++v) acc[tt][v] = fb;
    }
    #pragma unroll
    for (int kk = 0; kk < 16; ++kk) {
      int k = kk * 4 + hi * 2;
      v2f a; a.x = hbuf[p ^ 1][l16][k]; a.y = hbuf[p ^ 1][l16][k + 1];
      #pragma unroll
      for (int tt = 0; tt < 4; ++tt) {
        int col = wave * 64 + tt * 16 + l16;
        v2f b; b.x = fc_w[col * HIDDEN + k]; b.y = fc_w[col * HIDDEN + k + 1];
        acc[tt] = wmma_f32(a, b, acc[tt]);
      }
    }

    // ---- row max (xor-shuffles stay within 16-lane halves = C-tile rows) ----
    #pragma unroll
    for (int v = 0; v < 8; ++v) {
      float mx = acc[0][v];
      #pragma unroll
      for (int tt = 1; tt < 4; ++tt) mx = fmaxf(mx, acc[tt][v]);
      #pragma unroll
      for (int off = 8; off >= 1; off >>= 1)
        mx = fmaxf(mx, __shfl_xor(mx, off, 32));
      if (l16 == 0) redv[wave][v + hi * 8] = mx;
    }
    __syncthreads();
    if (tid < ROWS) {
      float mx = redv[0][tid];
      for (int w = 1; w < NW; ++w) mx = fmaxf(mx, redv[w][tid]);
      rowmax[tid] = mx;
    }
    __syncthreads();

    // ---- row sum(exp) -> log ----
    #pragma unroll
    for (int v = 0; v < 8; ++v) {
      float rm = rowmax[v + hi * 8];
      float s = 0.0f;
      #pragma unroll
      for (int tt = 0; tt < 4; ++tt) s += __expf(acc[tt][v] - rm);
      #pragma unroll
      for (int off = 8; off >= 1; off >>= 1)
        s += __shfl_xor(s, off, 32);
      if (l16 == 0) redv[wave][v + hi * 8] = s;
    }
    __syncthreads();
    if (tid < ROWS) {
      float s = 0.0f;
      for (int w = 0; w < NW; ++w) s += redv[w][tid];
      rowls[tid] = __logf(s);
    }
    __syncthreads();

    // ---- write log-probs + greedy argmax (first-index tie-break) ----
    #pragma unroll
    for (int v = 0; v < 8; ++v) {
      int m = v + hi * 8;
      float rm = rowmax[m], ls = rowls[m];
      float bestv = -INFINITY; int besti = 0;
      #pragma unroll
      for (int tt = 0; tt < 4; ++tt) {
        int col = wave * 64 + tt * 16 + l16;
        float lp = acc[tt][v] - rm - ls;
        out_logp[((size_t)(r0 + m) * MAXLEN + t) * VOCAB + col] = lp;
        if (lp > bestv) { bestv = lp; besti = col; }   // strict > keeps lowest col
      }
      #pragma unroll
      for (int off = 8; off >= 1; off >>= 1) {
        float ov = __shfl_xor(bestv, off, 32);
        int   oi = __shfl_xor(besti, off, 32);
        if (ov > bestv || (ov == bestv && oi < besti)) { bestv = ov; besti = oi; }
      }
      if (l16 == 0) { redv[wave][m] = bestv; redi[wave][m] = besti; }
    }
    __syncthreads();
    if (tid < ROWS) {
      float bv = redv[0][tid]; int bi2 = redi[0][tid];
      for (int w = 1; w < NW; ++w) {
        float ov = redv[w][tid]; int oi = redi[w][tid];
        if (ov > bv || (ov == bv && oi < bi2)) { bv = ov; bi2 = oi; }
      }
      ids[tid] = bi2;
    }
    p ^= 1;
    __syncthreads();
  }

  // ---- final hidden state ----
  for (int e = tid; e < ROWS * HIDDEN; e += 256) {
    int m = e >> 6, k = e & 63;
    out_h[(size_t)(r0 + m) * HIDDEN + k] = hbuf[p][m][k];
  }
}

extern "C" void kernel_launch(void* const* d_in, const int* in_sizes, int n_in,
                              void* d_out, int out_size, void* d_ws, size_t ws_size,
                              hipStream_t stream) {
  const float* enc_h = (const float*)d_in[0];
  const float* embed = (const float*)d_in[1];
  const float* w_ih  = (const float*)d_in[2];
  const float* w_hh  = (const float*)d_in[3];
  const float* b_ih  = (const float*)d_in[4];
  const float* b_hh  = (const float*)d_in[5];
  const float* fc_w  = (const float*)d_in[6];
  const float* fc_b  = (const float*)d_in[7];

  float* out      = (float*)d_out;
  float* out_logp = out;                                     // [B,T,V]
  float* out_h    = out + (size_t)BATCH * MAXLEN * VOCAB;    // [1,B,H]

  dim3 grid(BATCH / ROWS);   // 512 workgroups
  dim3 block(256);           // 8 wave32s
  hipLaunchKernelGGL(Decoder_40742059770033_kernel, grid, block, 0, stream,
                     enc_h, embed, w_ih, w_hh, b_ih, b_hh, fc_w, fc_b,
                     out_logp, out_h);
}